// CustomMLPLayer_5514738008077
// MI455X (gfx1250) — compile-verified
//
#include <hip/hip_runtime.h>
#include <stdint.h>

typedef float v2f __attribute__((ext_vector_type(2)));
typedef float v8f __attribute__((ext_vector_type(8)));
typedef unsigned int uint32x4 __attribute__((ext_vector_type(4)));
typedef int int32x4 __attribute__((ext_vector_type(4)));
typedef int int32x8 __attribute__((ext_vector_type(8)));

#define SEQ     2048
#define HIDDEN  4096
#define INTER   11008
#define BASE_NUM  2201
#define EXTRA_NUM 2201
#define TOKEN_K   2201

// ---------------- GEMM: C[s,h] = sum_k X[s,k] * W[h,k] ----------------
// Block tile 64(s) x 128(h), K-chunk 32, 8 waves; each wave owns a 32x32 tile
// (4 f32 WMMA accumulators). Tiles DMA'd into LDS by the Tensor Data Mover
// with hardware row padding (32 DWORDs data + 4 DWORDs pad = stride 36).
#define BM 64
#define BN 128
#define BK 32
#define LDK 36   // padded LDS row stride in floats (TDM pad: interval 32 dw, amount 4 dw)

// Build D# group 1 (256b) for a 2D f32 tensor: dim0 = INTER (K), row-major,
// tile = 32 x tile_dim1, LDS padding enabled.
__device__ __forceinline__ int32x8 tdm_desc_g1(unsigned tensor_dim1, unsigned tile_dim1) {
  int32x8 g;
  g[0] = (2 << 16)        // data_size = 4B
       | (1 << 20)        // pad_enable
       | (4 << 22)        // pad_interval: 2<<4 = 32 DWORDs
       | (3 << 25);       // pad_amount: 3+1 = 4 DWORDs
  g[1] = (int)(((unsigned)INTER & 0xFFFFu) << 16);              // tensor_dim0[15:0]
  g[2] = (int)(((unsigned)INTER >> 16) | ((tensor_dim1 & 0xFFFFu) << 16));
  g[3] = (int)((tensor_dim1 >> 16) | (32u << 16));              // tile_dim0 = 32
  g[4] = (int)tile_dim1;                                        // tile_dim1; tile_dim2 = 0
  g[5] = (int)INTER;                                            // tensor_dim0_stride lo
  g[6] = 0;                                                     // stride hi / dim1_stride
  g[7] = 0;
  return g;
}

// Issue TENSOR_LOAD_TO_LDS for one tile (2D -> groups 2/3 zero-filled).
__device__ __forceinline__ void tdm_load(unsigned lds_off, unsigned long long gaddr,
                                         int32x8 g1) {
  uint32x4 g0;
  g0[0] = 1u;                                                   // count = 1 valid D#
  g0[1] = lds_off;                                              // LDS byte address
  g0[2] = (unsigned)gaddr;                                      // global_addr[31:0]
  g0[3] = ((unsigned)(gaddr >> 32) & 0x01FFFFFFu) | (2u << 30); // addr[56:32] | type=2
  int32x4 gz = {0, 0, 0, 0};
#if __clang_major__ >= 23
  int32x8 gz8 = {};
  __builtin_amdgcn_tensor_load_to_lds(g0, g1, gz, gz, gz8, 0);
#else
  __builtin_amdgcn_tensor_load_to_lds(g0, g1, gz, gz, 0);
#endif
}

__global__ __launch_bounds__(256)
void gemm_wmma_f32(const float* __restrict__ X, const float* __restrict__ W,
                   float* __restrict__ C) {
  __shared__ float Abuf[2][BM * LDK];   // 2 x  9216 B
  __shared__ float Bbuf[2][BN * LDK];   // 2 x 18432 B  (total 55296 B)

  const int tid  = threadIdx.x;
  const int lane = tid & 31;
  const int wave = tid >> 5;

  const int s0 = blockIdx.y * BM;
  const int h0 = blockIdx.x * BN;

  // 2x4 wave grid of 32x32 tiles
  const int wm = (wave & 1) * 32;
  const int wn = (wave >> 1) * 32;

  const int lm = lane & 15;          // M/N index within fragment
  const int lk = (lane >> 4) * 2;    // lanes 0-15: k+0..1, lanes 16-31: k+2..3

  const unsigned long long gA = (unsigned long long)(uintptr_t)X + (size_t)s0 * INTER * 4u;
  const unsigned long long gB = (unsigned long long)(uintptr_t)W + (size_t)h0 * INTER * 4u;

  const int32x8 dA = tdm_desc_g1(SEQ, BM);     // tile 32 x 64
  const int32x8 dB = tdm_desc_g1(HIDDEN, BN);  // tile 32 x 128

  v8f c00 = {}, c01 = {}, c10 = {}, c11 = {};

  const int NCHUNK = INTER / BK;     // 344

  if (wave == 0) {                   // prologue: DMA chunk 0 into buffer 0
    tdm_load((unsigned)(uintptr_t)&Abuf[0][0], gA, dA);
    tdm_load((unsigned)(uintptr_t)&Bbuf[0][0], gB, dB);
  }

  for (int c = 0; c < NCHUNK; ++c) {
    const int cur = c & 1;
    if (wave == 0) {
      if (c + 1 < NCHUNK) {
        const unsigned long long koff = (unsigned long long)(c + 1) * BK * 4u;
        tdm_load((unsigned)(uintptr_t)&Abuf[cur ^ 1][0], gA + koff, dA);
        tdm_load((unsigned)(uintptr_t)&Bbuf[cur ^ 1][0], gB + koff, dB);
        __builtin_amdgcn_s_wait_tensorcnt(2);  // in-order: current buffer landed
      } else {
        __builtin_amdgcn_s_wait_tensorcnt(0);
      }
    }
    __syncthreads();                 // DMA'd tiles visible to all waves

    const float* As = &Abuf[cur][0];
    const float* Bs = &Bbuf[cur][0];

#pragma unroll
    for (int ks = 0; ks < BK; ks += 4) {
      v2f a0 = *(const v2f*)(&As[(wm + lm) * LDK + ks + lk]);
      v2f a1 = *(const v2f*)(&As[(wm + 16 + lm) * LDK + ks + lk]);
      v2f b0 = *(const v2f*)(&Bs[(wn + lm) * LDK + ks + lk]);
      v2f b1 = *(const v2f*)(&Bs[(wn + 16 + lm) * LDK + ks + lk]);
      c00 = __builtin_amdgcn_wmma_f32_16x16x4_f32(false, a0, false, b0, (short)0, c00, false, false);
      c01 = __builtin_amdgcn_wmma_f32_16x16x4_f32(false, a0, false, b1, (short)0, c01, false, false);
      c10 = __builtin_amdgcn_wmma_f32_16x16x4_f32(false, a1, false, b0, (short)0, c10, false, false);
      c11 = __builtin_amdgcn_wmma_f32_16x16x4_f32(false, a1, false, b1, (short)0, c11, false, false);
    }
    __syncthreads();                 // all waves done reading before next DMA overwrite
  }

  // C/D layout: VGPR j -> M = j (lanes 0-15) or j+8 (lanes 16-31), N = lane&15
  const int col0 = h0 + wn + lm;
  const int row0 = s0 + wm + ((lane >> 4) ? 8 : 0);
#pragma unroll
  for (int j = 0; j < 8; ++j) {
    C[(size_t)(row0 + j) * HIDDEN + col0]           = c00[j];
    C[(size_t)(row0 + j) * HIDDEN + col0 + 16]      = c01[j];
    C[(size_t)(row0 + 16 + j) * HIDDEN + col0]      = c10[j];
    C[(size_t)(row0 + 16 + j) * HIDDEN + col0 + 16] = c11[j];
  }
}

// ---------------- index pipeline ----------------

__global__ void zero_counts(unsigned* counts) {
  int i = blockIdx.x * blockDim.x + threadIdx.x;
  if (i < INTER) counts[i] = 0u;
}

__device__ __forceinline__ unsigned f2ord(float f) {
  unsigned b = __float_as_uint(f);
  return (b & 0x80000000u) ? ~b : (b | 0x80000000u);
}

// One block per token row: radix-select the TOKEN_K-th largest (exact on the
// ordered-uint keys), then bump counts[i] for every element >= threshold.
__global__ __launch_bounds__(256)
void topk_count(const float* __restrict__ X, unsigned* __restrict__ counts) {
  __shared__ unsigned keys[INTER];     // 44032 B row staged in LDS
  __shared__ unsigned hist[256];
  __shared__ unsigned s_prefix, s_remaining;

  const int s   = blockIdx.x;
  const int tid = threadIdx.x;
  const float* row = X + (size_t)s * INTER;

  for (int i = tid; i < INTER; i += 256) keys[i] = f2ord(row[i]);   // 43 iters exact
  if (tid == 0) { s_prefix = 0u; s_remaining = TOKEN_K; }
  __syncthreads();

  for (int shift = 24; shift >= 0; shift -= 8) {
    hist[tid] = 0u;
    __syncthreads();
    const unsigned mask   = (shift == 24) ? 0u : (0xFFFFFFFFu << (shift + 8));
    const unsigned prefix = s_prefix;
    for (int i = tid; i < INTER; i += 256) {
      unsigned u = keys[i];
      if ((u & mask) == prefix) atomicAdd(&hist[(u >> shift) & 255u], 1u);
    }
    __syncthreads();
    if (tid == 0) {
      unsigned rem = s_remaining, cum = 0u;
      for (int b = 255; b >= 0; --b) {
        unsigned c = hist[b];
        if (cum + c >= rem) {
          s_prefix    = prefix | ((unsigned)b << shift);
          s_remaining = rem - cum;
          break;
        }
        cum += c;
      }
    }
    __syncthreads();
  }

  const unsigned thr = s_prefix;   // exact value of the k-th largest key
  for (int i = tid; i < INTER; i += 256)
    if (keys[i] >= thr) atomicAdd(&counts[i], 1u);
}

// Single block: among ids in [BASE_NUM, INTER), take top EXTRA_NUM ordered by
// (count desc, id asc) via bitonic sort of packed keys, then emit indices_all.
#define RANK_N 16384
__global__ __launch_bounds__(1024)
void rank_emit(const unsigned* __restrict__ counts, int* __restrict__ out_idx) {
  __shared__ unsigned keys[RANK_N];   // 64 KB
  const int tid = threadIdx.x;

  for (int i = tid; i < RANK_N; i += 1024) {
    const int id = BASE_NUM + i;
    unsigned key = 0u;
    if (id < INTER) key = (counts[id] << 14) | (unsigned)(INTER - 1 - id);
    keys[i] = key;
  }
  __syncthreads();

  for (int size = 2; size <= RANK_N; size <<= 1) {
    for (int stride = size >> 1; stride > 0; stride >>= 1) {
      for (int i = tid; i < RANK_N / 2; i += 1024) {   // 8 iters exact
        const int lo = ((i / stride) * (stride * 2)) + (i % stride);
        const int hi = lo + stride;
        const bool desc = ((lo & size) == 0);          // final merge: descending
        unsigned a = keys[lo], b = keys[hi];
        if (desc ? (a < b) : (a > b)) { keys[lo] = b; keys[hi] = a; }
      }
      __syncthreads();
    }
  }

  for (int j = tid; j < BASE_NUM; j += 1024) out_idx[j] = j;
  for (int j = tid; j < EXTRA_NUM; j += 1024)
    out_idx[BASE_NUM + j] = (INTER - 1) - (int)(keys[j] & 0x3FFFu);
}

// ---------------- launcher ----------------
extern "C" void kernel_launch(void* const* d_in, const int* in_sizes, int n_in,
                              void* d_out, int out_size, void* d_ws, size_t ws_size,
                              hipStream_t stream) {
  const float* x = (const float*)d_in[0];   // [1, SEQ, INTER]
  const float* w = (const float*)d_in[1];   // [HIDDEN, INTER]

  float* true_value = (float*)d_out;                                // SEQ*HIDDEN f32
  int*   indices    = (int*)((float*)d_out + (size_t)SEQ * HIDDEN); // 4402 i32
  unsigned* counts  = (unsigned*)d_ws;                              // INTER u32

  zero_counts<<<(INTER + 255) / 256, 256, 0, stream>>>(counts);

  dim3 grid(HIDDEN / BN, SEQ / BM);  // 32 x 32
  gemm_wmma_f32<<<grid, 256, 0, stream>>>(x, w, true_value);

  topk_count<<<SEQ, 256, 0, stream>>>(x, counts);
  rank_emit<<<1, 1024, 0, stream>>>(counts, indices);
}